// SparseGraphAttention_68453188764119
// MI455X (gfx1250) — compile-verified
//
#include <hip/hip_runtime.h>
#include <cstdint>
#include <cstddef>

// Sparse graph attention for MI455X (gfx1250, wave32).
// Dense projections run on V_WMMA_F32_16X16X4_F32 (exact fp32, reference-matching).
// Edge passes are coalesced gathers + native f32/int hardware atomics.
// All device-side address math is 32-bit (buffers < 2^31 bytes) so loads lower
// to GVS mode: SGPR base + 32-bit voffset + immediate, no per-iter u64 adds.

constexpr int NN = 100000;   // nodes
constexpr int NE = 1600000;  // edges
constexpr int DH = 128;      // model dim
constexpr int NH = 8;        // heads
constexpr int HD = 16;       // head dim

typedef __attribute__((ext_vector_type(2))) float v2f;
typedef __attribute__((ext_vector_type(8))) float v8f;

#define DEV __device__ __forceinline__

// Order-preserving integer encoding for float atomic max (native int atomics,
// no CAS loop). Requires destination initialized to -INF.
DEV void atomic_max_f32(float* addr, float val) {
  if (val >= 0.0f) atomicMax(reinterpret_cast<int*>(addr), __float_as_int(val));
  else             atomicMin(reinterpret_cast<unsigned int*>(addr), __float_as_uint(val));
}

DEV void atomic_add_f32(float* addr, float val) {
  unsafeAtomicAdd(addr, val);  // -> global_atomic_add_f32
}

// ---------------------------------------------------------------------------
// Fused K/V projection: Kb = kv @ Wk^T + bk ; Vb = kv @ Wv^T + bv
// One wave -> one 16x16 output tile, A-fragment shared by both GEMMs.
// grid = NN/16 blocks of 256 threads (8 waves = 8 column tiles of 16).
// ---------------------------------------------------------------------------
__global__ void __launch_bounds__(256)
proj_kv_wmma(const float* __restrict__ kv,
             const float* __restrict__ Wk, const float* __restrict__ bk,
             const float* __restrict__ Wv, const float* __restrict__ bv,
             float* __restrict__ Kb, float* __restrict__ Vb)
{
  const int m0   = blockIdx.x << 4;
  const int n0   = ((int)threadIdx.x >> 5) << 4;
  const int lane = (int)threadIdx.x & 31;
  const int hi   = lane >> 4;   // 0: K pair {0,1}, 1: K pair {2,3}
  const int lr   = lane & 15;

  // 32-bit element offsets (per-lane bases, loop adds immediates only)
  const int aOff = (m0 + lr) * DH + (hi << 1);   // A: 16x4 fp32 frag
  const int wOff = (n0 + lr) * DH + (hi << 1);   // B: rows of W (C = A*W^T)

  v8f accK = {0.f,0.f,0.f,0.f,0.f,0.f,0.f,0.f};
  v8f accV = {0.f,0.f,0.f,0.f,0.f,0.f,0.f,0.f};
#pragma unroll 8
  for (int k = 0; k < DH; k += 4) {
    v2f a  = *(const v2f*)(kv + aOff + k);
    v2f bK = *(const v2f*)(Wk + wOff + k);
    v2f bV = *(const v2f*)(Wv + wOff + k);
    accK = __builtin_amdgcn_wmma_f32_16x16x4_f32(false, a, false, bK,
                                                 (short)0, accK, false, false);
    accV = __builtin_amdgcn_wmma_f32_16x16x4_f32(false, a, false, bV,
                                                 (short)0, accV, false, false);
  }

  const float biasK = bk[n0 + lr];
  const float biasV = bv[n0 + lr];
  // C/D layout: VGPR g -> M = g + 8*hi, N = lr
  const int oBase = (m0 + (hi << 3)) * DH + n0 + lr;
#pragma unroll
  for (int g = 0; g < 8; ++g) {
    Kb[oBase + g * DH] = accK[g] + biasK;
    Vb[oBase + g * DH] = accV[g] + biasV;
  }
}

// ---------------------------------------------------------------------------
// Output projection: out = aggr @ Wo^T + bo
// ---------------------------------------------------------------------------
__global__ void __launch_bounds__(256)
out_proj_wmma(const float* __restrict__ A,
              const float* __restrict__ Wo, const float* __restrict__ bo,
              float* __restrict__ out)
{
  const int m0   = blockIdx.x << 4;
  const int n0   = ((int)threadIdx.x >> 5) << 4;
  const int lane = (int)threadIdx.x & 31;
  const int hi   = lane >> 4;
  const int lr   = lane & 15;

  const int aOff = (m0 + lr) * DH + (hi << 1);
  const int wOff = (n0 + lr) * DH + (hi << 1);

  v8f acc = {0.f,0.f,0.f,0.f,0.f,0.f,0.f,0.f};
#pragma unroll 8
  for (int k = 0; k < DH; k += 4) {
    v2f a = *(const v2f*)(A  + aOff + k);
    v2f b = *(const v2f*)(Wo + wOff + k);
    acc = __builtin_amdgcn_wmma_f32_16x16x4_f32(false, a, false, b,
                                                (short)0, acc, false, false);
  }
  const float bias = bo[n0 + lr];
  const int oBase = (m0 + (hi << 3)) * DH + n0 + lr;
#pragma unroll
  for (int g = 0; g < 8; ++g) {
    out[oBase + g * DH] = acc[g] + bias;
  }
}

// ---------------------------------------------------------------------------
// Init: seg-max = -inf, seg-sum = 0  (NN*NH elements each)
// ---------------------------------------------------------------------------
__global__ void __launch_bounds__(256)
init_seg(float* __restrict__ m, float* __restrict__ s)
{
  const int t = blockIdx.x * 256 + (int)threadIdx.x;
  if (t < NN * NH) { m[t] = -INFINITY; s[t] = 0.0f; }
}

__global__ void __launch_bounds__(256)
zero_buf(float* __restrict__ p, int n)
{
  const int t = blockIdx.x * 256 + (int)threadIdx.x;
  if (t < n) p[t] = 0.0f;
}

// ---------------------------------------------------------------------------
// Pass A: scores[e,h] = (K[src,h,:] . Q[dst,h,:]) / sqrt(16); seg-max via atomics.
// Thread = (edge, head); 8 consecutive lanes stream one contiguous 512B row.
// ---------------------------------------------------------------------------
__global__ void __launch_bounds__(256)
edge_scores(const float* __restrict__ Kb, const float* __restrict__ q,
            const int* __restrict__ ei,
            float* __restrict__ scores, float* __restrict__ segmax)
{
  const int t = blockIdx.x * 256 + (int)threadIdx.x;   // NE*NH = 12.8M < 2^31
  const int e = t >> 3;
  const int h = t & 7;
  const int s = ei[e];        // edge_index[0][e] = src
  const int d = ei[NE + e];   // edge_index[1][e] = dst

  const float4* kp = (const float4*)(Kb + s * DH + h * HD);
  const float4* qp = (const float4*)(q  + d * DH + h * HD);
  float acc = 0.0f;
#pragma unroll
  for (int i = 0; i < 4; ++i) {
    const float4 a = kp[i], b = qp[i];
    acc += a.x * b.x + a.y * b.y + a.z * b.z + a.w * b.w;
  }
  const float sc = acc * 0.25f;   // 1/sqrt(HD)
  scores[t] = sc;
  atomic_max_f32(&segmax[d * NH + h], sc);
}

// ---------------------------------------------------------------------------
// Pass B: ex = exp(score - max[dst]); seg-sum += ex (in-place over scores)
// ---------------------------------------------------------------------------
__global__ void __launch_bounds__(256)
edge_softmax(const int* __restrict__ ei, const float* __restrict__ segmax,
             float* __restrict__ scores, float* __restrict__ segsum)
{
  const int t = blockIdx.x * 256 + (int)threadIdx.x;
  const int e = t >> 3;
  const int h = t & 7;
  const int d = ei[NE + e];
  const float ex = __expf(scores[t] - segmax[d * NH + h]);
  scores[t] = ex;
  atomic_add_f32(&segsum[d * NH + h], ex);
}

// ---------------------------------------------------------------------------
// Pass C: aggr[dst,h,:] += V[src,h,:] * (ex / (sum[dst,h] + 1e-16))
// ---------------------------------------------------------------------------
__global__ void __launch_bounds__(256)
edge_aggregate(const int* __restrict__ ei, const float* __restrict__ Vb,
               const float* __restrict__ scores, const float* __restrict__ segsum,
               float* __restrict__ aggr)
{
  const int t = blockIdx.x * 256 + (int)threadIdx.x;
  const int e = t >> 3;
  const int h = t & 7;
  const int s = ei[e];
  const int d = ei[NE + e];
  const float attn = scores[t] / (segsum[d * NH + h] + 1e-16f);

  const float4* vp = (const float4*)(Vb + s * DH + h * HD);
  float* op = aggr + d * DH + h * HD;
#pragma unroll
  for (int i = 0; i < 4; ++i) {
    const float4 v = vp[i];
    atomic_add_f32(op + 4 * i + 0, v.x * attn);
    atomic_add_f32(op + 4 * i + 1, v.y * attn);
    atomic_add_f32(op + 4 * i + 2, v.z * attn);
    atomic_add_f32(op + 4 * i + 3, v.w * attn);
  }
}

// ---------------------------------------------------------------------------
// Launch. Workspace layout (floats), total 40M floats = 160 MB:
//   [0)            Kb   NN*DH   -- reused as aggr after Pass A (K is dead then)
//   [NN*DH)        Vb   NN*DH
//   [2*NN*DH)      sc   NE*NH   (scores, then exp(scores) in place)
//   [.. + NE*NH)   m    NN*NH   (segment max)
//   [.. + NN*NH)   ssum NN*NH   (segment sum)
// ---------------------------------------------------------------------------
extern "C" void kernel_launch(void* const* d_in, const int* in_sizes, int n_in,
                              void* d_out, int out_size, void* d_ws, size_t ws_size,
                              hipStream_t stream)
{
  const float* q  = (const float*)d_in[0];
  const float* kv = (const float*)d_in[1];
  const int*   ei = (const int*)  d_in[2];
  const float* Wk = (const float*)d_in[3];
  const float* bk = (const float*)d_in[4];
  const float* Wv = (const float*)d_in[5];
  const float* bv = (const float*)d_in[6];
  const float* Wo = (const float*)d_in[7];
  const float* bo = (const float*)d_in[8];
  float* out = (float*)d_out;

  float* Kb   = (float*)d_ws;
  float* Vb   = Kb + (size_t)NN * DH;
  float* sc   = Vb + (size_t)NN * DH;
  float* segm = sc + (size_t)NE * NH;
  float* segs = segm + (size_t)NN * NH;
  float* aggr = Kb;  // alias: K buffer is dead after edge_scores

  const int TPB = 256;
  const int gProj = NN / 16;                 // 6250
  const int gEdge = (NE * NH) / TPB;         // 50000
  const int gSeg  = (NN * NH + TPB - 1) / TPB;
  const int gNode = (NN * DH) / TPB;         // 50000

  init_seg      <<<gSeg,  TPB, 0, stream>>>(segm, segs);
  proj_kv_wmma  <<<gProj, TPB, 0, stream>>>(kv, Wk, bk, Wv, bv, Kb, Vb);
  edge_scores   <<<gEdge, TPB, 0, stream>>>(Kb, q, ei, sc, segm);
  zero_buf      <<<gNode, TPB, 0, stream>>>(aggr, NN * DH);   // after scores: Kb dead
  edge_softmax  <<<gEdge, TPB, 0, stream>>>(ei, segm, sc, segs);
  edge_aggregate<<<gEdge, TPB, 0, stream>>>(ei, Vb, sc, segs, aggr);
  out_proj_wmma <<<gProj, TPB, 0, stream>>>(aggr, Wo, bo, out);
}